// MSDeformableAttention_18262200942810
// MI455X (gfx1250) — compile-verified
//
#include <hip/hip_runtime.h>
#include <hip/hip_bf16.h>
#include <math.h>

// Problem constants (fixed by the reference)
#define BS    8
#define LQ    1000
#define DIM   256
#define NH    8
#define NL    4
#define NP    4
#define CC    32          // channels per head = DIM/NH
#define LEN_V 8500        // 80*80 + 40*40 + 20*20 + 10*10

typedef float        v2f __attribute__((ext_vector_type(2)));
typedef float        v8f __attribute__((ext_vector_type(8)));
typedef unsigned int v4u __attribute__((ext_vector_type(4)));
typedef int          v8i __attribute__((ext_vector_type(8)));
typedef int          v4i __attribute__((ext_vector_type(4)));

#if __has_builtin(__builtin_amdgcn_tensor_load_to_lds) && \
    __has_builtin(__builtin_amdgcn_s_wait_tensorcnt)
#define USE_TDM 1
#else
#define USE_TDM 0
#endif

// ---------------------------------------------------------------------------
// GEMM: C[M,NN] = A[M,KK] * B[KK,NN] + bias[NN]   (fp32, row-major)
//
// Block = 2 waves = 32 M-rows; each wave owns a 16(M) x 64(N) tile.
// grid.x = N tiles: the N-tiles sharing the same A rows are dispatch-
// adjacent, so their TDM loads of the same A slice hit L2 -> ~1x HBM on A.
//
// A path (the 70MB HBM stream): ONE Tensor Data Mover descriptor stages the
// block's [32 x KK] A-slice (32KB) into LDS (tensor_load_to_lds +
// s_wait_tensorcnt). The A fragment (a.x,a.y)=A[m,k],A[m,k+1] is contiguous,
// so it reads back as a single aligned ds_load_b64 - no shuffle moves.
// B path (<=256KB, L2-resident): direct global_load_b32 with immediate
// offsets (clean clauses). K fully unrolled: 256 V_WMMA_F32_16X16X4_F32 per
// wave with zero per-iteration address arithmetic.
//
// Lane layouts per CDNA5 ISA 7.12.2 (wave32):
//  A 16x4 : lanes 0-15 -> M=lane, v0=K0,v1=K1 ; lanes 16-31 -> M=lane-16, K2/K3
//  B 4x16 : v0: lanes 0-15 = row K0, lanes 16-31 = row K2 ; v1: K1 / K3
//  C 16x16: VGPR r: lanes 0-15 -> M=r, N=lane ; lanes 16-31 -> M=r+8, N=lane-16
// ---------------------------------------------------------------------------
template<int NN, int KK>
__global__ __launch_bounds__(64)
void gemm_wmma(const float* __restrict__ A, const float* __restrict__ B,
               const float* __restrict__ bias, float* __restrict__ C, int M)
{
    const int wave   = threadIdx.x >> 5;
    const int tile_n = blockIdx.x * 64;
    const int tile_m = blockIdx.y * 32 + wave * 16;
    const int lane   = threadIdx.x & 31;
    const int half   = lane >> 4;        // 0: lanes 0-15, 1: lanes 16-31
    const int l16    = lane & 15;
    const int koff   = half * 2;         // K offset owned by this half-wave

    v8f acc[4] = {};
    const float* bp = B + (size_t)koff * NN + tile_n + l16;

#if USE_TDM
    // ---- stage A[blockIdx.y*32 : +32, 0:KK] into LDS with the TDM --------
    __shared__ float asm_[32 * KK];      // 32KB, rows contiguous
    if (threadIdx.x < 32) {              // one wave issues the descriptor
        const unsigned long long ga =
            (unsigned long long)(A + (size_t)blockIdx.y * 32 * KK);
        v4u g0;
        g0.x = 1u;                                   // count=1 (valid D#)
        g0.y = 0u;                                   // lds_addr = 0 (asm_ base)
        g0.z = (unsigned)ga;                         // global_addr[31:0]
        g0.w = (unsigned)((ga >> 32) & 0x01FFFFFFu)  // global_addr[56:32]
             | 0x80000000u;                          // type=2 ("image")
        v8i g1;
        g1[0] = 0x20000;                             // data_size=2 (4 bytes)
        g1[1] = (int)((unsigned)(KK & 0xFFFF) << 16);        // tensor_dim0 lo
        g1[2] = (int)(((unsigned)M & 0xFFFFu) << 16);        // dim0 hi | tensor_dim1 lo
        g1[3] = (int)(((unsigned)M >> 16) | ((unsigned)KK << 16)); // dim1 hi | tile_dim0=KK
        g1[4] = 32;                                  // tile_dim1=32 | tile_dim2=0
        g1[5] = KK;                                  // tensor_dim0_stride[31:0]
        g1[6] = 0;                                   // stride0 hi | stride1 lo
        g1[7] = 0;                                   // stride1 hi
        const v4i gz4 = {0, 0, 0, 0};                // groups 2/3 unused (2D)
        const v8i gz8 = {0, 0, 0, 0, 0, 0, 0, 0};
        __builtin_amdgcn_tensor_load_to_lds(g0, g1, gz4, gz4, gz8, 0);
        __builtin_amdgcn_s_wait_tensorcnt(0);
    }
    __syncthreads();
    // A fragment base for this lane: row (wave*16 + l16), col koff
    const float* al = asm_ + (size_t)(wave * 16 + l16) * KK + koff;
#pragma unroll
    for (int k = 0; k < KK; k += 4) {
        const v2f a = *(const v2f*)(al + k);         // one ds_load_b64
#pragma unroll
        for (int t = 0; t < 4; ++t) {
            v2f b;
            b.x = bp[(size_t)(k    ) * NN + 16 * t]; // global, imm offset
            b.y = bp[(size_t)(k + 1) * NN + 16 * t];
            acc[t] = __builtin_amdgcn_wmma_f32_16x16x4_f32(
                         false, a, false, b, (short)0, acc[t], false, false);
        }
    }
#else
    const float* ap = A + (size_t)(tile_m + l16) * KK + koff;
#pragma unroll
    for (int k = 0; k < KK; k += 4) {
        const v2f a = *(const v2f*)(ap + k);
#pragma unroll
        for (int t = 0; t < 4; ++t) {
            v2f b;
            b.x = bp[(size_t)(k    ) * NN + 16 * t];
            b.y = bp[(size_t)(k + 1) * NN + 16 * t];
            acc[t] = __builtin_amdgcn_wmma_f32_16x16x4_f32(
                         false, a, false, b, (short)0, acc[t], false, false);
        }
    }
#endif

#pragma unroll
    for (int t = 0; t < 4; ++t) {
        const int   col = tile_n + 16 * t + l16;
        const float bv  = bias ? bias[col] : 0.f;
#pragma unroll
        for (int r = 0; r < 8; ++r) {
            const int row = tile_m + r + half * 8;
            C[(size_t)row * NN + col] = acc[t][r] + bv;
        }
    }
}

// ---------------------------------------------------------------------------
// Softmax over the last-16 (NL*NP) axis, one thread per (b, q, head) row.
// ---------------------------------------------------------------------------
__global__ void softmax16(float* __restrict__ a, int rows)
{
    const int i = blockIdx.x * blockDim.x + threadIdx.x;
    if (i >= rows) return;
    float* p = a + (size_t)i * 16;
    float v[16];
    float m = -3.402823466e38f;
#pragma unroll
    for (int j = 0; j < 16; ++j) { v[j] = p[j]; m = fmaxf(m, v[j]); }
    float s = 0.f;
#pragma unroll
    for (int j = 0; j < 16; ++j) { v[j] = __expf(v[j] - m); s += v[j]; }
    const float inv = 1.f / s;
#pragma unroll
    for (int j = 0; j < 16; ++j) p[j] = v[j] * inv;
}

// ---------------------------------------------------------------------------
// Bilinear deformable sampling + attention-weighted sum.
// grid = BS*LQ blocks; block = 256 threads = 8 heads x 32 channels.
// Channel c maps to lane (t & 31): each tap is 32 consecutive floats ->
// one coalesced 128B load per tap per head (L2-resident, 192MB L2).
// v layout: [b, idx, head, c] ; out layout: [b, q, head*CC + c].
// ---------------------------------------------------------------------------
__global__ __launch_bounds__(256)
void ms_sample(const float* __restrict__ v,    const float* __restrict__ refp,
               const float* __restrict__ off,  const float* __restrict__ attn,
               const int*   __restrict__ shapes, const int* __restrict__ starts,
               float* __restrict__ out)
{
    const int bq = blockIdx.x;          // 0 .. BS*LQ-1
    const int b  = bq / LQ;
    const int t  = threadIdx.x;
    const int h  = t >> 5;              // head
    const int c  = t & 31;              // channel within head

    const float* offp = off  + (size_t)bq * DIM            + h * (NL * NP * 2);
    const float* ap   = attn + (size_t)bq * (NH * NL * NP) + h * (NL * NP);
    const float* rp   = refp + (size_t)bq * (NL * 2);

    float acc = 0.f;
#pragma unroll
    for (int l = 0; l < NL; ++l) {
        const int H  = shapes[2 * l];
        const int W  = shapes[2 * l + 1];
        const int st = starts[l];
        const float rx = rp[2 * l];
        const float ry = rp[2 * l + 1];
        const float* vb = v + (((size_t)b * LEN_V + st) * NH + h) * CC + c;
#pragma unroll
        for (int p = 0; p < NP; ++p) {
            const float ox = offp[(l * NP + p) * 2 + 0];
            const float oy = offp[(l * NP + p) * 2 + 1];
            // loc = ref + off/(W,H); x = loc_x*W - 0.5 (align_corners=False)
            const float x  = (rx + ox / (float)W) * (float)W - 0.5f;
            const float y  = (ry + oy / (float)H) * (float)H - 0.5f;
            const float xf = floorf(x), yf = floorf(y);
            const float fx = x - xf,    fy = y - yf;
            const int   x0 = (int)xf,   y0 = (int)yf;
            float s = 0.f;
#pragma unroll
            for (int dy = 0; dy < 2; ++dy)
#pragma unroll
            for (int dx = 0; dx < 2; ++dx) {
                const int xx = x0 + dx, yy = y0 + dy;
                if (xx >= 0 && xx < W && yy >= 0 && yy < H) {
                    const float wt = (dx ? fx : 1.f - fx) * (dy ? fy : 1.f - fy);
                    s += wt * vb[(size_t)(yy * W + xx) * (NH * CC)];
                }
            }
            acc += ap[l * NP + p] * s;
        }
    }
    out[(size_t)bq * DIM + h * CC + c] = acc;
}

// ---------------------------------------------------------------------------
extern "C" void kernel_launch(void* const* d_in, const int* in_sizes, int n_in,
                              void* d_out, int out_size, void* d_ws, size_t ws_size,
                              hipStream_t stream)
{
    const float* query  = (const float*)d_in[0];
    const float* refp   = (const float*)d_in[1];
    const float* value  = (const float*)d_in[2];
    const float* w_off  = (const float*)d_in[3];
    const float* b_off  = (const float*)d_in[4];
    const float* w_attn = (const float*)d_in[5];
    const float* b_attn = (const float*)d_in[6];
    const float* w_v    = (const float*)d_in[7];
    const float* b_v    = (const float*)d_in[8];
    const float* w_o    = (const float*)d_in[9];
    const float* b_o    = (const float*)d_in[10];
    const int*   shapes = (const int*)d_in[11];
    const int*   starts = (const int*)d_in[12];

    // Workspace layout (fp32): v_buf | off_buf | attn_buf | tmp_out  (~90 MB)
    float* ws       = (float*)d_ws;
    float* v_buf    = ws;                                          // 68000*256
    float* off_buf  = v_buf    + (size_t)BS * LEN_V * DIM;         //  8000*256
    float* attn_buf = off_buf  + (size_t)BS * LQ * DIM;            //  8000*128
    float* tmp_out  = attn_buf + (size_t)BS * LQ * (NH * NL * NP); //  8000*256

    const dim3 gblk(64);   // 2 waves / 32 M-rows per block

    // 1) v = value @ w_v + b_v          [68000 x 256]
    gemm_wmma<DIM, DIM><<<dim3(DIM / 64, BS * LEN_V / 32), gblk, 0, stream>>>(
        value, w_v, b_v, v_buf, BS * LEN_V);

    // 2) off = query @ w_off + b_off    [8000 x 256]
    gemm_wmma<DIM, DIM><<<dim3(DIM / 64, BS * LQ / 32), gblk, 0, stream>>>(
        query, w_off, b_off, off_buf, BS * LQ);

    // 3) attn logits = query @ w_attn + b_attn   [8000 x 128]
    gemm_wmma<NH * NL * NP, DIM>
        <<<dim3((NH * NL * NP) / 64, BS * LQ / 32), gblk, 0, stream>>>(
        query, w_attn, b_attn, attn_buf, BS * LQ);

    // 4) softmax over NL*NP=16 per (b,q,head)
    softmax16<<<(BS * LQ * NH + 255) / 256, 256, 0, stream>>>(
        attn_buf, BS * LQ * NH);

    // 5) deformable bilinear sampling + weighted sum -> tmp_out [8000 x 256]
    ms_sample<<<BS * LQ, 256, 0, stream>>>(
        v_buf, refp, off_buf, attn_buf, shapes, starts, tmp_out);

    // 6) out = tmp_out @ w_o + b_o -> d_out  [8000 x 256]
    gemm_wmma<DIM, DIM><<<dim3(DIM / 64, BS * LQ / 32), gblk, 0, stream>>>(
        tmp_out, w_o, b_o, (float*)d_out, BS * LQ);
}